// ModelWithEdgeFeatures_23837068493269
// MI455X (gfx1250) — compile-verified
//
#include <hip/hip_runtime.h>
#include <hip/hip_bf16.h>

typedef __attribute__((ext_vector_type(2))) float v2f;
typedef __attribute__((ext_vector_type(8))) float v8f;

#define N_NODES  50000
#define N_EDGES  800000
#define NP       50048        // N padded to multiple of 128
#define OUTW     128          // hidden width, all layers
#define EDGE_CH  16
#define G_NUM    128
#define MLP_HID  256
#define NCLS     10
#define BN_EPS   1e-5f

// ---------------------------------------------------------------- WMMA helper
__device__ __forceinline__ v8f wmma4(v2f a, v2f b, v8f c) {
  // V_WMMA_F32_16X16X4_F32 : D = A(16x4,f32) x B(4x16,f32) + C(16x16,f32)
  return __builtin_amdgcn_wmma_f32_16x16x4_f32(
      /*neg_a=*/false, a, /*neg_b=*/false, b,
      /*c_mod=*/(short)0, c, /*reuse_a=*/false, /*reuse_b=*/false);
}

// ---------------------------------------------------------------- utilities
__global__ void zero_f(float* __restrict__ p, int n) {
  int t = blockIdx.x * blockDim.x + threadIdx.x;
  if (t < n) p[t] = 0.0f;
}

__global__ void copy_f(float* __restrict__ d, const float* __restrict__ s, int n) {
  int t = blockIdx.x * blockDim.x + threadIdx.x;
  if (t < n) d[t] = s[t];
}

// deg[i] = #incoming edges; sea[i] = sum of edge_attr over edges with dst==i
__global__ void deg_sea_kernel(const int* __restrict__ ei, const float* __restrict__ ea,
                               float* __restrict__ deg, float* __restrict__ sea) {
  int t = blockIdx.x * blockDim.x + threadIdx.x;
  if (t >= N_EDGES * EDGE_CH) return;
  int e = t >> 4, j = t & 15;
  int dst = ei[N_EDGES + e];
  atomicAdd(&sea[(size_t)dst * EDGE_CH + j], ea[(size_t)e * EDGE_CH + j]);
  if (j == 0) atomicAdd(&deg[dst], 1.0f);
}

// ah[dst] += h[src]  (float4-vectorized channels)
__global__ void scatter_edges(const int* __restrict__ ei, const float* __restrict__ h,
                              float* __restrict__ ah, int prev) {
  int per = prev >> 2;
  int t = blockIdx.x * blockDim.x + threadIdx.x;
  if (t >= N_EDGES * per) return;
  int e = t / per, q = t - e * per;
  int src = ei[e];
  int dst = ei[N_EDGES + e];
  const float4 hv = *(const float4*)(h + (size_t)src * prev + q * 4);
  float* ap = ah + (size_t)dst * prev + q * 4;
  atomicAdd(ap + 0, hv.x);
  atomicAdd(ap + 1, hv.y);
  atomicAdd(ap + 2, hv.z);
  atomicAdd(ap + 3, hv.w);
}

// ---------------------------------------------------------------- fused layer GEMM
// out = relu(BN(relu( [(deg+1)*h | ah+h | sea+1] @ W + (deg+1)*b )))
// Block: 256 thr = 8 waves; each wave -> 16 rows x 128 cols (8 WMMA tiles).
template <int PREV>
__global__ __launch_bounds__(256) void gemm_layer(
    const float* __restrict__ h,    // [NP, PREV]
    const float* __restrict__ ahp,  // [NP, PREV]  (= ah + h)
    const float* __restrict__ sea,  // [NP, 16]
    const float* __restrict__ deg,  // [NP]
    const float* __restrict__ W,    // [2*PREV+16, 128]
    const float* __restrict__ bias,
    const float* __restrict__ bng, const float* __restrict__ bnb,
    const float* __restrict__ bnm, const float* __restrict__ bnv,
    float* __restrict__ out)        // [NP, 128]
{
  const int lane = threadIdx.x & 31;
  const int wav  = threadIdx.x >> 5;
  const int lm   = lane & 15;
  const int kh   = lane >> 4;                 // K-half select (0/1)
  const int row0 = blockIdx.x * 128 + wav * 16;
  const int arow = row0 + lm;                 // A row this lane supplies

  v8f acc[8] = {};

  const float srow = deg[arow] + 1.0f;

  // --- segment 1: (deg+1)*h @ W[0:PREV) ---
  {
    const float* __restrict__ A = h + (size_t)arow * PREV;
#pragma unroll 4
    for (int k0 = 0; k0 < PREV; k0 += 4) {
      const int ka = k0 + 2 * kh;
      v2f a; a.x = A[ka] * srow; a.y = A[ka + 1] * srow;
      const float* wp = W + (size_t)ka * OUTW + lm;
#pragma unroll
      for (int j = 0; j < 8; ++j) {
        v2f b; b.x = wp[j * 16]; b.y = wp[OUTW + j * 16];
        acc[j] = wmma4(a, b, acc[j]);
      }
    }
  }
  // --- segment 2: (ah+h) @ W[PREV:2*PREV) ---
  {
    const float* __restrict__ A  = ahp + (size_t)arow * PREV;
    const float* __restrict__ W2 = W + (size_t)PREV * OUTW;
#pragma unroll 4
    for (int k0 = 0; k0 < PREV; k0 += 4) {
      const int ka = k0 + 2 * kh;
      v2f a; a.x = A[ka]; a.y = A[ka + 1];
      const float* wp = W2 + (size_t)ka * OUTW + lm;
#pragma unroll
      for (int j = 0; j < 8; ++j) {
        v2f b; b.x = wp[j * 16]; b.y = wp[OUTW + j * 16];
        acc[j] = wmma4(a, b, acc[j]);
      }
    }
  }
  // --- segment 3: (sum_ea + 1) @ W[2*PREV:2*PREV+16) ---
  {
    const float* __restrict__ A  = sea + (size_t)arow * EDGE_CH;
    const float* __restrict__ W3 = W + (size_t)(2 * PREV) * OUTW;
#pragma unroll
    for (int k0 = 0; k0 < EDGE_CH; k0 += 4) {
      const int ka = k0 + 2 * kh;
      v2f a; a.x = A[ka] + 1.0f; a.y = A[ka + 1] + 1.0f;
      const float* wp = W3 + (size_t)ka * OUTW + lm;
#pragma unroll
      for (int j = 0; j < 8; ++j) {
        v2f b; b.x = wp[j * 16]; b.y = wp[OUTW + j * 16];
        acc[j] = wmma4(a, b, acc[j]);
      }
    }
  }

  // --- epilogue: + (deg+1)*b, relu, BN(eval), relu ---
  float s2[8];
#pragma unroll
  for (int vv = 0; vv < 8; ++vv) {
    int row = row0 + vv + kh * 8;
    s2[vv] = deg[row] + 1.0f;           // pad rows zeroed -> safe
  }
#pragma unroll
  for (int j = 0; j < 8; ++j) {
    const int col = j * 16 + lm;
    const float bc = bias[col];
    const float g  = bng[col];
    const float bo = bnb[col];
    const float mu = bnm[col];
    const float rs = 1.0f / sqrtf(bnv[col] + BN_EPS);
#pragma unroll
    for (int vv = 0; vv < 8; ++vv) {
      const int row = row0 + vv + kh * 8;
      if (row < N_NODES) {
        float val = acc[j][vv] + s2[vv] * bc;
        val = fmaxf(val, 0.0f);
        val = (val - mu) * rs * g + bo;
        val = fmaxf(val, 0.0f);
        out[(size_t)row * OUTW + col] = val;
      }
    }
  }
}

// ---------------------------------------------------------------- pooling + head
__global__ void pool_kernel(const float* __restrict__ h, const int* __restrict__ batch,
                            float* __restrict__ pooled) {
  int t = blockIdx.x * blockDim.x + threadIdx.x;
  if (t >= N_NODES * OUTW) return;
  int i = t >> 7, c = t & 127;
  atomicAdd(&pooled[(size_t)batch[i] * OUTW + c], h[(size_t)i * OUTW + c]);
}

__global__ void head1(const float* __restrict__ pooled, const float* __restrict__ fw1,
                      const float* __restrict__ fb1, float* __restrict__ hid) {
  int t = blockIdx.x * blockDim.x + threadIdx.x;
  if (t >= G_NUM * MLP_HID) return;
  int g = t / MLP_HID, o = t - g * MLP_HID;
  float s = fb1[o];
  for (int k = 0; k < OUTW; ++k) s += pooled[g * OUTW + k] * fw1[k * MLP_HID + o];
  hid[t] = fmaxf(s, 0.0f);
}

__global__ void head2(const float* __restrict__ hid, const float* __restrict__ fw2,
                      const float* __restrict__ fb2, float* __restrict__ outp) {
  int t = blockIdx.x * blockDim.x + threadIdx.x;
  if (t >= G_NUM * NCLS) return;
  int g = t / NCLS, o = t - g * NCLS;
  float s = fb2[o];
  for (int k = 0; k < MLP_HID; ++k) s += hid[g * MLP_HID + k] * fw2[k * NCLS + o];
  outp[t] = s;
}

// ---------------------------------------------------------------- launch
extern "C" void kernel_launch(void* const* d_in, const int* in_sizes, int n_in,
                              void* d_out, int out_size, void* d_ws, size_t ws_size,
                              hipStream_t stream) {
  if (n_in < 26) return;
  const float* x     = (const float*)d_in[0];
  const int*   ei    = (const int*)d_in[1];
  const float* ea    = (const float*)d_in[2];
  const int*   batch = (const int*)d_in[3];
  const float* W[3]  = {(const float*)d_in[4], (const float*)d_in[5], (const float*)d_in[6]};
  const float* lb[3] = {(const float*)d_in[7], (const float*)d_in[8], (const float*)d_in[9]};
  const float* bg[3] = {(const float*)d_in[10], (const float*)d_in[11], (const float*)d_in[12]};
  const float* bb[3] = {(const float*)d_in[13], (const float*)d_in[14], (const float*)d_in[15]};
  const float* bm[3] = {(const float*)d_in[16], (const float*)d_in[17], (const float*)d_in[18]};
  const float* bv[3] = {(const float*)d_in[19], (const float*)d_in[20], (const float*)d_in[21]};
  const float* fw1 = (const float*)d_in[22];
  const float* fb1 = (const float*)d_in[23];
  const float* fw2 = (const float*)d_in[24];
  const float* fb2 = (const float*)d_in[25];
  float* outp = (float*)d_out;

  float* p   = (float*)d_ws;
  float* deg = p;     p += NP;                 // [NP]
  float* sea = p;     p += (size_t)NP * 16;    // [NP,16]  (contiguous after deg)
  float* ah  = p;     p += (size_t)NP * 128;   // scatter accumulator
  float* h1  = p;     p += (size_t)NP * 128;
  float* h2  = p;     p += (size_t)NP * 128;
  float* pooled = p;  p += (size_t)G_NUM * OUTW;
  float* hid = p;     p += (size_t)G_NUM * MLP_HID;
  size_t need = (size_t)(p - (float*)d_ws) * sizeof(float);
  if (ws_size < need) return;

  const dim3 blk(256);
  auto grid1 = [](long long n) { return dim3((unsigned)((n + 255) / 256)); };

  // once-per-call invariants: degrees + edge-attr sums (zero pads included)
  zero_f<<<grid1((long long)NP * 17), blk, 0, stream>>>(deg, NP * 17);
  deg_sea_kernel<<<grid1((long long)N_EDGES * 16), blk, 0, stream>>>(ei, ea, deg, sea);

  // ---- layer 1 (prev = 64) ----
  copy_f<<<grid1((long long)N_NODES * 64), blk, 0, stream>>>(ah, x, N_NODES * 64);
  scatter_edges<<<grid1((long long)N_EDGES * 16), blk, 0, stream>>>(ei, x, ah, 64);
  gemm_layer<64><<<NP / 128, blk, 0, stream>>>(x, ah, sea, deg, W[0], lb[0],
                                               bg[0], bb[0], bm[0], bv[0], h1);
  // ---- layer 2 (prev = 128) ----
  copy_f<<<grid1((long long)N_NODES * 128), blk, 0, stream>>>(ah, h1, N_NODES * 128);
  scatter_edges<<<grid1((long long)N_EDGES * 32), blk, 0, stream>>>(ei, h1, ah, 128);
  gemm_layer<128><<<NP / 128, blk, 0, stream>>>(h1, ah, sea, deg, W[1], lb[1],
                                                bg[1], bb[1], bm[1], bv[1], h2);
  // ---- layer 3 (prev = 128), output reuses h1 ----
  copy_f<<<grid1((long long)N_NODES * 128), blk, 0, stream>>>(ah, h2, N_NODES * 128);
  scatter_edges<<<grid1((long long)N_EDGES * 32), blk, 0, stream>>>(ei, h2, ah, 128);
  gemm_layer<128><<<NP / 128, blk, 0, stream>>>(h2, ah, sea, deg, W[2], lb[2],
                                                bg[2], bb[2], bm[2], bv[2], h1);

  // ---- global add-pool + MLP head ----
  zero_f<<<grid1((long long)G_NUM * OUTW), blk, 0, stream>>>(pooled, G_NUM * OUTW);
  pool_kernel<<<grid1((long long)N_NODES * OUTW), blk, 0, stream>>>(h1, batch, pooled);
  head1<<<grid1((long long)G_NUM * MLP_HID), blk, 0, stream>>>(pooled, fw1, fb1, hid);
  head2<<<grid1((long long)G_NUM * NCLS), blk, 0, stream>>>(hid, fw2, fb2, outp);
}